// GConv_43868795961412
// MI455X (gfx1250) — compile-verified
//
#include <hip/hip_runtime.h>
#include <hip/hip_bf16.h>

// ---------------------------------------------------------------------------
// Mixture-of-GCNs forward for MI455X (gfx1250, wave32, WMMA).
// GEMMs use V_WMMA_F32_16X16X4_F32 (full fp32 fidelity -- GEMM is not the
// bottleneck; the edge scatter / BN sweeps are, and they live in L2).
// ---------------------------------------------------------------------------

typedef __attribute__((ext_vector_type(2))) float v2f;
typedef __attribute__((ext_vector_type(8))) float v8f;

#define DCH 128          // D_IN = D_HID = D_EMB
#define GMIX 4
#define BN_EPS 1e-5f

// ------------------------------ GEMM ---------------------------------------
// C[M x 128] = A[M x 128] * W[128 x 128]
// block = 256 threads (8 waves); each wave computes one 16-row M tile x all
// 128 output columns (8 N-tiles). W is staged transposed into LDS so the B
// fragment (K,K+1) is one ds_load_b64 (compiler merges pairs into
// ds_load_2addr_stride64_b64).
__global__ __launch_bounds__(256) void gcn_gemm128_wmma(
    const float* __restrict__ A, const float* __restrict__ W,
    float* __restrict__ C, int M) {
  __shared__ float Wt[DCH * DCH];   // Wt[n*128 + k] = W[k*128 + n]  (64 KB)

  const int t = threadIdx.x;
  // Cooperative transpose-stage of W: 16 float4 per thread, coalesced reads.
  #pragma unroll
  for (int i = 0; i < 16; ++i) {
    int f  = t + 256 * i;          // float4 index 0..4095
    int k  = f >> 5;               // row of W (32 float4 per 128-float row)
    int nb = (f & 31) << 2;        // starting column
    float4 w4 = ((const float4*)W)[f];
    Wt[(nb + 0) * DCH + k] = w4.x;
    Wt[(nb + 1) * DCH + k] = w4.y;
    Wt[(nb + 2) * DCH + k] = w4.z;
    Wt[(nb + 3) * DCH + k] = w4.w;
  }
  __syncthreads();

  const int wave = t >> 5;
  const int lane = t & 31;
  const int mt   = blockIdx.x * 8 + wave;      // 16-row tile index
  if (mt * 16 >= M) return;                    // whole-wave predicate: EXEC all-1

  // A fragment addressing per ISA 32-bit 16x4 A layout:
  //   lanes 0-15 hold M=lane,K={k,k+1}; lanes 16-31 hold M=lane-16,K={k+2,k+3}
  const int mrow  = mt * 16 + (lane & 15);
  const int khalf = (lane >> 4) * 2;           // 0 or 2
  const float* arow = A + (size_t)mrow * DCH + khalf;
  // B fragment base per ISA 4x16 B layout (rows striped across lanes):
  const float* bbase = &Wt[(lane & 15) * DCH + khalf];

  v8f acc[8];
  #pragma unroll
  for (int nt = 0; nt < 8; ++nt) acc[nt] = {};

  for (int kk = 0; kk < DCH; kk += 4) {
    v2f a = *(const v2f*)(arow + kk);
    // WGP-scope prefetch (locality=3): warm the WGP cache for the strided
    // per-lane A stream one cache line ahead.
    __builtin_prefetch(arow + kk + 32, 0, 3);
    #pragma unroll
    for (int nt = 0; nt < 8; ++nt) {
      v2f b = *(const v2f*)(bbase + nt * 16 * DCH + kk);   // LDS b64 fragment
      acc[nt] = __builtin_amdgcn_wmma_f32_16x16x4_f32(
          /*neg_a=*/false, a, /*neg_b=*/false, b,
          /*c_mod=*/(short)0, acc[nt], /*reuse_a=*/false, /*reuse_b=*/false);
    }
  }

  // C/D layout: VGPR v -> row = v + 8*(lane>=16), col = lane & 15.
  const int crow0 = mt * 16 + (lane >> 4) * 8;
  const int ccol  = lane & 15;
  #pragma unroll
  for (int nt = 0; nt < 8; ++nt) {
    #pragma unroll
    for (int v = 0; v < 8; ++v) {
      C[(size_t)(crow0 + v) * DCH + nt * 16 + ccol] = acc[nt][v];
    }
  }
}

// --------------------------- graph / BN kernels -----------------------------

__global__ void k_fill(float* p, float v, int n) {
  int i = blockIdx.x * blockDim.x + threadIdx.x;
  if (i < n) p[i] = v;
}

__global__ void k_deg_accum(const int* __restrict__ col, float* deg, int E) {
  int e = blockIdx.x * blockDim.x + threadIdx.x;
  if (e < E) atomicAdd(&deg[col[e]], 1.0f);
}

__global__ void k_dinv(float* dinv, int n) {   // in place: deg -> deg^-1/2
  int i = blockIdx.x * blockDim.x + threadIdx.x;
  if (i < n) { float d = dinv[i]; dinv[i] = d > 0.0f ? rsqrtf(d) : 0.0f; }
}

__global__ void k_softmax4(const float* __restrict__ me, float* mix, int n) {
  int i = blockIdx.x * blockDim.x + threadIdx.x;
  if (i >= n) return;
  float a0 = me[4 * i + 0], a1 = me[4 * i + 1], a2 = me[4 * i + 2], a3 = me[4 * i + 3];
  float m = fmaxf(fmaxf(a0, a1), fmaxf(a2, a3));
  float e0 = __expf(a0 - m), e1 = __expf(a1 - m), e2 = __expf(a2 - m), e3 = __expf(a3 - m);
  float inv = 1.0f / (e0 + e1 + e2 + e3);
  mix[4 * i + 0] = e0 * inv; mix[4 * i + 1] = e1 * inv;
  mix[4 * i + 2] = e2 * inv; mix[4 * i + 3] = e3 * inv;
}

// z[i][c..c+3] = dinv[i]^2 * h[i][c..c+3] + bias[c..c+3]   (vectorized b128)
__global__ void k_zinit(const float* __restrict__ h, const float* __restrict__ dinv,
                        const float* __restrict__ bias, float* z, int n32) {
  int q = blockIdx.x * blockDim.x + threadIdx.x;   // float4 index
  if (q >= n32) return;
  int i = q >> 5;            // 32 float4 per row
  int c4 = q & 31;
  float di = dinv[i];
  float d2 = di * di;
  float4 hv = ((const float4*)h)[q];
  float4 bv = ((const float4*)bias)[c4];
  float4 r;
  r.x = d2 * hv.x + bv.x; r.y = d2 * hv.y + bv.y;
  r.z = d2 * hv.z + bv.z; r.w = d2 * hv.w + bv.w;
  ((float4*)z)[q] = r;
}

// scatter-add: z[col] += dinv[row]*dinv[col] * h[row]; one wave per edge.
__global__ void k_edge_scatter(const int* __restrict__ row, const int* __restrict__ col,
                               const float* __restrict__ dinv,
                               const float* __restrict__ h, float* z, int E) {
  int t = blockIdx.x * blockDim.x + threadIdx.x;
  int e = t >> 5;
  if (e >= E) return;
  int lane = t & 31;
  int r = row[e], c = col[e];
  float nrm = dinv[r] * dinv[c];
  const float* hr = h + (size_t)r * DCH;
  float* zc = z + (size_t)c * DCH;
  #pragma unroll
  for (int j = 0; j < 4; ++j) {
    int ch = lane + 32 * j;
    atomicAdd(&zc[ch], nrm * hr[ch]);
  }
}

// per-column partial sums; blockDim = 128 (thread = column), 256 rows/block.
__global__ __launch_bounds__(128) void k_bn_stats(const float* __restrict__ z,
                                                  float* stats, int n) {
  int c = threadIdx.x;
  int r0 = blockIdx.x * 256;
  int r1 = min(r0 + 256, n);
  float s = 0.0f, s2 = 0.0f;
  for (int r = r0; r < r1; ++r) {
    float v = z[(size_t)r * DCH + c];
    s += v; s2 += v * v;
  }
  atomicAdd(&stats[c], s);
  atomicAdd(&stats[DCH + c], s2);
}

__global__ __launch_bounds__(128) void k_bn_finalize(const float* __restrict__ stats,
                                                     const float* __restrict__ gamma,
                                                     const float* __restrict__ beta,
                                                     float* scale, float* shift, float invN) {
  int c = threadIdx.x;
  float mu  = stats[c] * invN;
  float var = stats[DCH + c] * invN - mu * mu;
  float sc  = gamma[c] * rsqrtf(var + BN_EPS);
  scale[c] = sc;
  shift[c] = beta[c] - mu * sc;
}

// in-place BN + PReLU, 4 channels/thread (b128 traffic)
__global__ void k_bn_prelu(float* z, const float* __restrict__ scale,
                           const float* __restrict__ shift,
                           const float* __restrict__ alpha_p, int n32) {
  int q = blockIdx.x * blockDim.x + threadIdx.x;
  if (q >= n32) return;
  int c4 = q & 31;
  float a = alpha_p[0];
  float4 zv = ((float4*)z)[q];
  float4 sc = ((const float4*)scale)[c4];
  float4 sh = ((const float4*)shift)[c4];
  float4 r;
  r.x = zv.x * sc.x + sh.x; r.x = r.x > 0.0f ? r.x : a * r.x;
  r.y = zv.y * sc.y + sh.y; r.y = r.y > 0.0f ? r.y : a * r.y;
  r.z = zv.z * sc.z + sh.z; r.z = r.z > 0.0f ? r.z : a * r.z;
  r.w = zv.w * sc.w + sh.w; r.w = r.w > 0.0f ? r.w : a * r.w;
  ((float4*)z)[q] = r;
}

// out += mix[:,g] * prelu(bn(z)), 4 channels/thread
__global__ void k_final(const float* __restrict__ z, const float* __restrict__ scale,
                        const float* __restrict__ shift, const float* __restrict__ alpha_p,
                        const float* __restrict__ mix, float* out, int n32, int g) {
  int q = blockIdx.x * blockDim.x + threadIdx.x;
  if (q >= n32) return;
  int i = q >> 5;
  int c4 = q & 31;
  float a = alpha_p[0];
  float w = mix[4 * i + g];
  float4 zv = ((const float4*)z)[q];
  float4 sc = ((const float4*)scale)[c4];
  float4 sh = ((const float4*)shift)[c4];
  float4 ov = ((float4*)out)[q];
  float v;
  v = zv.x * sc.x + sh.x; v = v > 0.0f ? v : a * v; ov.x += w * v;
  v = zv.y * sc.y + sh.y; v = v > 0.0f ? v : a * v; ov.y += w * v;
  v = zv.z * sc.z + sh.z; v = v > 0.0f ? v : a * v; ov.z += w * v;
  v = zv.w * sc.w + sh.w; v = v > 0.0f ? v : a * v; ov.w += w * v;
  ((float4*)out)[q] = ov;
}

// ------------------------------ launch --------------------------------------

extern "C" void kernel_launch(void* const* d_in, const int* in_sizes, int n_in,
                              void* d_out, int out_size, void* d_ws, size_t ws_size,
                              hipStream_t stream) {
  const float* x      = (const float*)d_in[0];
  const int*   edges  = (const int*)d_in[1];
  const float* W1     = (const float*)d_in[2];
  const float* b1     = (const float*)d_in[3];
  const float* W2     = (const float*)d_in[4];
  const float* b2     = (const float*)d_in[5];
  const float* gamma1 = (const float*)d_in[6];
  const float* beta1  = (const float*)d_in[7];
  const float* gamma2 = (const float*)d_in[8];
  const float* beta2  = (const float*)d_in[9];
  const float* alpha  = (const float*)d_in[10];
  const float* membed = (const float*)d_in[11];
  float* out = (float*)d_out;

  const int N  = in_sizes[0] / DCH;
  const int E  = in_sizes[1] / 2;
  const int n128 = N * DCH;
  const int n32  = n128 / 4;           // float4 count per feature map
  const float invN = 1.0f / (float)N;

  const int* erow = edges;       // edge_index[0]
  const int* ecol = edges + E;   // edge_index[1]

  // workspace layout (floats)
  float* ws    = (float*)d_ws;
  float* h     = ws;                 // N*128
  float* z     = h + (size_t)n128;   // N*128
  float* dinv  = z + (size_t)n128;   // N
  float* mix   = dinv + N;           // N*4
  float* stats = mix + (size_t)N * 4;// 256 (sum | sumsq)
  float* scale = stats + 256;        // 128
  float* shift = scale + DCH;        // 128

  const int B = 256;
  dim3 blk(B);
  dim3 g_n128((n128 + B - 1) / B);
  dim3 g_n32((n32 + B - 1) / B);
  dim3 g_n((N + B - 1) / B);
  dim3 g_e((E + B - 1) / B);
  dim3 g_edge32(((size_t)E * 32 + B - 1) / B);
  dim3 g_gemm((N / 16 + 7) / 8);          // N multiple of 16 (50000 = 16*3125)
  dim3 g_bn((N + 255) / 256);

  // out = 0 (d_out is poisoned by the harness)
  k_fill<<<g_n128, blk, 0, stream>>>(out, 0.0f, n128);

  // symmetric GCN norm: deg (self-loop = 1) -> dinv
  k_fill<<<g_n, blk, 0, stream>>>(dinv, 1.0f, N);
  k_deg_accum<<<g_e, blk, 0, stream>>>(ecol, dinv, E);
  k_dinv<<<g_n, blk, 0, stream>>>(dinv, N);

  // softmax mixture weights
  k_softmax4<<<g_n, blk, 0, stream>>>(membed, mix, N);

  for (int g = 0; g < GMIX; ++g) {
    // ---- layer 1 ----
    gcn_gemm128_wmma<<<g_gemm, blk, 0, stream>>>(x, W1 + (size_t)g * DCH * DCH, h, N);
    k_zinit<<<g_n32, blk, 0, stream>>>(h, dinv, b1 + g * DCH, z, n32);
    k_edge_scatter<<<g_edge32, blk, 0, stream>>>(erow, ecol, dinv, h, z, E);
    k_fill<<<dim3(1), blk, 0, stream>>>(stats, 0.0f, 256);
    k_bn_stats<<<g_bn, dim3(128), 0, stream>>>(z, stats, N);
    k_bn_finalize<<<dim3(1), dim3(128), 0, stream>>>(stats, gamma1 + g * DCH,
                                                     beta1 + g * DCH, scale, shift, invN);
    k_bn_prelu<<<g_n32, blk, 0, stream>>>(z, scale, shift, alpha, n32);

    // ---- layer 2 ----
    gcn_gemm128_wmma<<<g_gemm, blk, 0, stream>>>(z, W2 + (size_t)g * DCH * DCH, h, N);
    k_zinit<<<g_n32, blk, 0, stream>>>(h, dinv, b2 + g * DCH, z, n32);
    k_edge_scatter<<<g_edge32, blk, 0, stream>>>(erow, ecol, dinv, h, z, E);
    k_fill<<<dim3(1), blk, 0, stream>>>(stats, 0.0f, 256);
    k_bn_stats<<<g_bn, dim3(128), 0, stream>>>(z, stats, N);
    k_bn_finalize<<<dim3(1), dim3(128), 0, stream>>>(stats, gamma2 + g * DCH,
                                                     beta2 + g * DCH, scale, shift, invN);
    k_final<<<g_n32, blk, 0, stream>>>(z, scale, shift, alpha, mix, out, n32, g);
  }
}